// TopKPoolNet_51788715655371
// MI455X (gfx1250) — compile-verified
//
#include <hip/hip_runtime.h>
#include <hip/hip_bf16.h>

// ---------------- problem constants (from reference) ----------------
#define BB   64
#define NN   1024
#define EPER 16
#define EE   (BB * NN * EPER)   // 1,048,576 edges
#define F_IN 64
#define HH   128
#define KK1  820
#define KK2  656
#define KK3  8
#define MM0  (BB * NN)          // 65536
#define MM1  (BB * KK1)         // 52480
#define MM2  (BB * KK2)         // 41984
#define MM3  (BB * KK3)         // 512

typedef float v2f __attribute__((ext_vector_type(2)));
typedef float v8f __attribute__((ext_vector_type(8)));

__device__ __forceinline__ void atomAddF(float* p, float v) {
    __hip_atomic_fetch_add(p, v, __ATOMIC_RELAXED, __HIP_MEMORY_SCOPE_AGENT);
}

// ------------- edge gather + scatter-add (segment_sum) --------------
// thread -> (edge, 4-feature chunk); consecutive lanes read consecutive
// floats of one source row (coalesced), atomics go to the dst row.
__global__ void agg_edges(const float* __restrict__ x,
                          const int* __restrict__ src,
                          const int* __restrict__ dst,
                          const int* __restrict__ mask,   // nullptr => all valid
                          float* __restrict__ agg,
                          int nE, int F) {
    const int chunks = F >> 2;
    long long t = (long long)blockIdx.x * blockDim.x + threadIdx.x;
    long long total = (long long)nE * chunks;
    if (t >= total) return;
    int e = (int)(t / chunks);
    int c = (int)(t % chunks);
    if (mask && mask[e] == 0) return;
    int s = src[e], d = dst[e];
    const float4 v = *(const float4*)(x + (size_t)s * F + c * 4);
    float* ap = agg + (size_t)d * F + c * 4;
    atomAddF(ap + 0, v.x);
    atomAddF(ap + 1, v.y);
    atomAddF(ap + 2, v.z);
    atomAddF(ap + 3, v.w);
}

// ------- fused dual GEMM: out = relu(A1@W1 + A2@W2 + bias) ----------
// out is [M x 128]; A1,A2 are [M x K] row-major; W1,W2 are [K x 128].
// One wave per 16x16 output tile via V_WMMA_F32_16X16X4_F32 (exact fp32).
// A fragment (16x4): lane L holds M=L%16, K = 2*(L/16)+j in vgpr j.
// B fragment (4x16): lane L holds N=L%16, K = 2*(L/16)+j in vgpr j.
// C/D (16x16): vgpr r, lane L -> row r+8*(L/16), col L%16.
__global__ __launch_bounds__(256)
void gemm_dual_relu(const float* __restrict__ A1, const float* __restrict__ W1,
                    const float* __restrict__ A2, const float* __restrict__ W2,
                    const float* __restrict__ bias, float* __restrict__ out,
                    int K) {
    const int lane = threadIdx.x & 31;
    const int wv   = threadIdx.x >> 5;   // n-tile 0..7 (covers all 128 cols)
    const int mt   = blockIdx.x;         // m-tile
    const int mr   = lane & 15;
    const int kg   = lane >> 4;          // 0 or 1
    const int row  = mt * 16 + mr;
    const int col  = wv * 16 + mr;

    v8f acc = {};
    const float* a1p = A1 + (size_t)row * K;
    const float* a2p = A2 + (size_t)row * K;
    for (int k0 = 0; k0 < K; k0 += 4) {
        const int ka = k0 + 2 * kg;
        v2f a1 = *(const v2f*)(a1p + ka);
        v2f a2 = *(const v2f*)(a2p + ka);
        v2f b1; b1.x = W1[(size_t)ka * HH + col]; b1.y = W1[(size_t)(ka + 1) * HH + col];
        v2f b2; b2.x = W2[(size_t)ka * HH + col]; b2.y = W2[(size_t)(ka + 1) * HH + col];
        acc = __builtin_amdgcn_wmma_f32_16x16x4_f32(false, a1, false, b1,
                                                    (short)0, acc, false, false);
        acc = __builtin_amdgcn_wmma_f32_16x16x4_f32(false, a2, false, b2,
                                                    (short)0, acc, false, false);
    }
    const float bc = bias[col];
#pragma unroll
    for (int r = 0; r < 8; ++r) {
        int orow = mt * 16 + r + 8 * kg;
        float v = acc[r] + bc;
        out[(size_t)orow * HH + col] = v > 0.f ? v : 0.f;
    }
}

// --------- score = tanh(<x,p>/||p||), one wave per node row ----------
__global__ void score_rows(const float* __restrict__ x, const float* __restrict__ p,
                           float* __restrict__ score, int M) {
    int row  = blockIdx.x * (blockDim.x >> 5) + (threadIdx.x >> 5);
    int lane = threadIdx.x & 31;
    if (row >= M) return;
    float dot = 0.f, pp = 0.f;
#pragma unroll
    for (int j = 0; j < 4; ++j) {
        float pv = p[lane + 32 * j];
        dot += x[(size_t)row * HH + lane + 32 * j] * pv;
        pp  += pv * pv;
    }
#pragma unroll
    for (int off = 16; off > 0; off >>= 1) {
        dot += __shfl_xor(dot, off, 32);
        pp  += __shfl_xor(pp, off, 32);
    }
    if (lane == 0) score[row] = tanhf(dot / sqrtf(pp));
}

// -------- per-graph top-k via LDS bitonic sort (desc, idx tiebreak) --
__global__ __launch_bounds__(1024)
void topk_sort(const float* __restrict__ score, int n_per, int k,
               int* __restrict__ perm, float* __restrict__ kscore,
               int* __restrict__ inv) {
    __shared__ float s[1024];
    __shared__ int   id[1024];
    const int b = blockIdx.x;
    const int t = threadIdx.x;
    s[t]  = (t < n_per) ? score[b * n_per + t] : -3.402823466e38f;
    id[t] = t;
    __syncthreads();
    for (int size = 2; size <= 1024; size <<= 1) {
        for (int stride = size >> 1; stride > 0; stride >>= 1) {
            int partner = t ^ stride;
            if (partner > t) {
                float s0 = s[t], s1 = s[partner];
                int   i0 = id[t], i1 = id[partner];
                bool descending = ((t & size) == 0);
                bool inOrder = (s0 > s1) || (s0 == s1 && i0 < i1);
                if (inOrder != descending) {
                    s[t] = s1; s[partner] = s0;
                    id[t] = i1; id[partner] = i0;
                }
            }
            __syncthreads();
        }
    }
    if (t < k) {
        int old_g = b * n_per + id[t];
        int new_g = b * k + t;
        perm[new_g]   = old_g;
        kscore[new_g] = s[t];
        inv[old_g]    = new_g;   // inv pre-filled with -1
    }
}

// ------------- x_new[new] = x[perm[new]] * kept_score ----------------
__global__ void pool_gather(const float* __restrict__ x, const int* __restrict__ perm,
                            const float* __restrict__ kscore, float* __restrict__ out) {
    int row = blockIdx.x;
    int f   = threadIdx.x;   // 128 threads
    out[(size_t)row * HH + f] = x[(size_t)perm[row] * HH + f] * kscore[row];
}

// ----------------------- edge id remap -------------------------------
__global__ void edge_remap(const int* __restrict__ src, const int* __restrict__ dst,
                           const int* __restrict__ mask, const int* __restrict__ inv,
                           int* __restrict__ nsrc, int* __restrict__ ndst,
                           int* __restrict__ nmask) {
    int e = blockIdx.x * blockDim.x + threadIdx.x;
    if (e >= EE) return;
    int v = mask ? mask[e] : 1;
    int s = inv[src[e]], d = inv[dst[e]];
    v = v && (s >= 0) && (d >= 0);
    nsrc[e]  = v ? s : 0;
    ndst[e]  = v ? d : 0;
    nmask[e] = v;
}

// ------------------- log_softmax over 128 features -------------------
__global__ void log_softmax_rows(const float* __restrict__ x, float* __restrict__ out,
                                 int M) {
    int row  = blockIdx.x * (blockDim.x >> 5) + (threadIdx.x >> 5);
    int lane = threadIdx.x & 31;
    if (row >= M) return;
    float v[4], mx = -3.402823466e38f;
#pragma unroll
    for (int j = 0; j < 4; ++j) {
        v[j] = x[(size_t)row * HH + lane + 32 * j];
        mx = fmaxf(mx, v[j]);
    }
#pragma unroll
    for (int off = 16; off > 0; off >>= 1) mx = fmaxf(mx, __shfl_xor(mx, off, 32));
    float se = 0.f;
#pragma unroll
    for (int j = 0; j < 4; ++j) se += expf(v[j] - mx);
#pragma unroll
    for (int off = 16; off > 0; off >>= 1) se += __shfl_xor(se, off, 32);
    float lse = logf(se);
#pragma unroll
    for (int j = 0; j < 4; ++j)
        out[(size_t)row * HH + lane + 32 * j] = v[j] - mx - lse;
}

// ===================================================================
extern "C" void kernel_launch(void* const* d_in, const int* in_sizes, int n_in,
                              void* d_out, int out_size, void* d_ws, size_t ws_size,
                              hipStream_t stream) {
    const float* x   = (const float*)d_in[0];
    const int*   ei  = (const int*)d_in[1];    // [2, E]: src then dst
    const float* W1r = (const float*)d_in[3];
    const float* W1l = (const float*)d_in[4];
    const float* b1  = (const float*)d_in[5];
    const float* p1  = (const float*)d_in[6];
    const float* W2r = (const float*)d_in[7];
    const float* W2l = (const float*)d_in[8];
    const float* b2  = (const float*)d_in[9];
    const float* p2  = (const float*)d_in[10];
    const float* W3r = (const float*)d_in[11];
    const float* W3l = (const float*)d_in[12];
    const float* b3  = (const float*)d_in[13];
    const float* p3  = (const float*)d_in[14];
    float* out = (float*)d_out;

    // ---------------- workspace carve-out ----------------
    char* ws = (char*)d_ws;
    size_t off = 0;
    auto take = [&](size_t bytes) -> void* {
        void* p = (void*)(ws + off);
        off += (bytes + 255) & ~(size_t)255;
        return p;
    };
    float* agg    = (float*)take((size_t)MM1 * HH * 4);  // covers M0*F_IN too
    float* hbuf   = (float*)take((size_t)MM0 * HH * 4);
    float* xp     = (float*)take((size_t)MM1 * HH * 4);
    float* score  = (float*)take((size_t)MM0 * 4);
    int*   perm   = (int*)  take((size_t)MM1 * 4);
    float* kscore = (float*)take((size_t)MM1 * 4);
    int*   inv    = (int*)  take((size_t)MM0 * 4);
    int*   esA    = (int*)  take((size_t)EE * 4);
    int*   edA    = (int*)  take((size_t)EE * 4);
    int*   emA    = (int*)  take((size_t)EE * 4);
    int*   esB    = (int*)  take((size_t)EE * 4);
    int*   edB    = (int*)  take((size_t)EE * 4);
    int*   emB    = (int*)  take((size_t)EE * 4);
    float* xp3    = (float*)take((size_t)MM3 * HH * 4);

    const int* e_src = ei;
    const int* e_dst = ei + EE;

    // ===================== conv1 =====================
    hipMemsetAsync(agg, 0, (size_t)MM0 * F_IN * 4, stream);
    {
        long long tot = (long long)EE * (F_IN / 4);
        agg_edges<<<dim3((unsigned)((tot + 255) / 256)), 256, 0, stream>>>(
            x, e_src, e_dst, nullptr, agg, EE, F_IN);
    }
    gemm_dual_relu<<<MM0 / 16, 256, 0, stream>>>(agg, W1r, x, W1l, b1, hbuf, F_IN);

    // ===================== pool1 =====================
    score_rows<<<(MM0 + 7) / 8, 256, 0, stream>>>(hbuf, p1, score, MM0);
    hipMemsetAsync(inv, 0xFF, (size_t)MM0 * 4, stream);
    topk_sort<<<BB, 1024, 0, stream>>>(score, NN, KK1, perm, kscore, inv);
    pool_gather<<<MM1, HH, 0, stream>>>(hbuf, perm, kscore, xp);
    edge_remap<<<EE / 256, 256, 0, stream>>>(e_src, e_dst, nullptr, inv, esA, edA, emA);

    // ===================== conv2 =====================
    hipMemsetAsync(agg, 0, (size_t)MM1 * HH * 4, stream);
    {
        long long tot = (long long)EE * (HH / 4);
        agg_edges<<<dim3((unsigned)((tot + 255) / 256)), 256, 0, stream>>>(
            xp, esA, edA, emA, agg, EE, HH);
    }
    gemm_dual_relu<<<MM1 / 16, 256, 0, stream>>>(agg, W2r, xp, W2l, b2, hbuf, HH);

    // ===================== pool2 =====================
    score_rows<<<(MM1 + 7) / 8, 256, 0, stream>>>(hbuf, p2, score, MM1);
    hipMemsetAsync(inv, 0xFF, (size_t)MM1 * 4, stream);
    topk_sort<<<BB, 1024, 0, stream>>>(score, KK1, KK2, perm, kscore, inv);
    pool_gather<<<MM2, HH, 0, stream>>>(hbuf, perm, kscore, xp);
    edge_remap<<<EE / 256, 256, 0, stream>>>(esA, edA, emA, inv, esB, edB, emB);

    // ===================== conv3 =====================
    hipMemsetAsync(agg, 0, (size_t)MM2 * HH * 4, stream);
    {
        long long tot = (long long)EE * (HH / 4);
        agg_edges<<<dim3((unsigned)((tot + 255) / 256)), 256, 0, stream>>>(
            xp, esB, edB, emB, agg, EE, HH);
    }
    gemm_dual_relu<<<MM2 / 16, 256, 0, stream>>>(agg, W3r, xp, W3l, b3, hbuf, HH);

    // ===================== pool3 (k = 8) =====================
    score_rows<<<(MM2 + 7) / 8, 256, 0, stream>>>(hbuf, p3, score, MM2);
    hipMemsetAsync(inv, 0xFF, (size_t)MM2 * 4, stream);
    topk_sort<<<BB, 1024, 0, stream>>>(score, KK2, KK3, perm, kscore, inv);
    pool_gather<<<MM3, HH, 0, stream>>>(hbuf, perm, kscore, xp3);

    // ===================== log_softmax =====================
    log_softmax_rows<<<MM3 / 8, 256, 0, stream>>>(xp3, out, MM3);
}